// GreenKernel_3375844294937
// MI455X (gfx1250) — compile-verified
//
#include <hip/hip_runtime.h>
#include <math.h>

#define EPSF 1e-8f
#define PI_F 3.14159265358979323846f

constexpr int Nn = 2048, Dd = 64, Rr = 128, Kk = 6;

// workspace float offsets
constexpr int GAMMA_OFF = 0;      // 6
constexpr int IMP_OFF   = 8;      // 128
constexpr int WG_OFF    = 136;    // [2][64] gaussian weights
constexpr int INVS_OFF  = 264;    // [2][64] helmholtz 1/(sigma+eps)
constexpr int ISIG2_OFF = 392;    // [2][64] mexican 1/(sigma^2+eps)
constexpr int CK_OFF    = 520;    // [2][128] sum_d w*a^2
constexpr int SK_OFF    = 776;    // [2][2048] sum_d w*z^2
constexpr int BM_OFF    = 4872;   // [2][64][128] B matrices (-2*w*a)

typedef __attribute__((ext_vector_type(2))) float v2f;
typedef __attribute__((ext_vector_type(8))) float v8f;

__global__ void prep_const_kernel(const float* __restrict__ gate,
                                  const float* __restrict__ anch,
                                  const float* __restrict__ rimp,
                                  const float* __restrict__ lsig,
                                  const float* __restrict__ lgam,
                                  float* __restrict__ ws) {
  __shared__ float wgl[128];
  const int t = threadIdx.x;
  if (t == 0) {
    float m = lgam[0];
    for (int k = 1; k < Kk; ++k) m = fmaxf(m, lgam[k]);
    float e[Kk]; float s = 0.f;
    for (int k = 0; k < Kk; ++k) { e[k] = __expf(lgam[k] - m); s += e[k]; }
    float inv = 1.f / s;
    for (int k = 0; k < Kk; ++k) ws[GAMMA_OFF + k] = e[k] * inv;
  }
  if (t < 128) {
    ws[IMP_OFF + t] = 1.f / (1.f + __expf(-rimp[t]));
    const int k = t >> 6, d = t & 63;
    // gaussian (rows 0,1): w = -gate/(2*sigma^2 + eps)
    float sg = __expf(lsig[k * Dd + d]);
    float w = -gate[d] / (2.f * sg * sg + EPSF);
    wgl[t] = w;
    ws[WG_OFF + t] = w;
    // helmholtz (rows 2,3): 1/(sigma+eps)
    float sh = __expf(lsig[(2 + k) * Dd + d]);
    ws[INVS_OFF + t] = 1.f / (sh + EPSF);
    // mexican hat (rows 4,5): 1/(sigma^2+eps)
    float sm = __expf(lsig[(4 + k) * Dd + d]);
    ws[ISIG2_OFF + t] = 1.f / (sm * sm + EPSF);
  }
  __syncthreads();
  {
    // c_k[r] = sum_d w * a^2   (256 threads -> k in {0,1}, r in 0..127)
    const int k = t >> 7, r = t & 127;
    float s = 0.f;
    for (int d = 0; d < Dd; ++d) {
      float a = anch[r * Dd + d];
      s += wgl[k * 64 + d] * a * a;
    }
    ws[CK_OFF + k * 128 + r] = s;
  }
  for (int i = t; i < 2 * 64 * 128; i += 256) {
    const int k = i >> 13;
    const int d = (i >> 7) & 63;
    const int r = i & 127;
    ws[BM_OFF + i] = -2.f * wgl[k * 64 + d] * anch[r * Dd + d];
  }
}

__global__ void prep_sz_kernel(const float* __restrict__ z,
                               float* __restrict__ ws) {
  const int n = blockIdx.x * blockDim.x + threadIdx.x;
  if (n >= Nn) return;
  float s0 = 0.f, s1 = 0.f;
  for (int d = 0; d < Dd; ++d) {
    float zz = z[n * Dd + d];
    zz *= zz;
    s0 += ws[WG_OFF + d] * zz;
    s1 += ws[WG_OFF + 64 + d] * zz;
  }
  ws[SK_OFF + n] = s0;
  ws[SK_OFF + Nn + n] = s1;
}

// One 256-thread block = 8 wave32; wave w owns output tile
// n in [bx*128 + w*16, +16), r in [by*16, +16).
__global__ __launch_bounds__(256) void phi_main_kernel(
    const float* __restrict__ z, const float* __restrict__ gate,
    const float* __restrict__ anch, const float* __restrict__ ws,
    float* __restrict__ out) {
  __shared__ float zs[128 * 68];   // z tile, rows padded to 68 floats
  __shared__ float as[16 * 68];    // anchor tile
  __shared__ float Bsh[2 * 64 * 16];
  __shared__ float gateL[64];
  __shared__ float invsL[128];
  __shared__ float isgL[128];
  __shared__ float ssL[256];
  __shared__ float ccL[32];
  __shared__ float impL[16];
  __shared__ float gamL[8];

  const int t  = threadIdx.x;
  const int nb = blockIdx.x;   // 0..15 : 128 n-rows each
  const int rb = blockIdx.y;   // 0..7  : 16 r-cols each

  for (int i = t; i < 128 * 64; i += 256) {
    const int n = i >> 6, d = i & 63;
    zs[n * 68 + d] = z[(nb * 128 + n) * Dd + d];
  }
  for (int i = t; i < 16 * 64; i += 256) {
    const int r = i >> 6, d = i & 63;
    as[r * 68 + d] = anch[(rb * 16 + r) * Dd + d];
  }
  for (int i = t; i < 2048; i += 256) {
    const int k = i >> 10, d = (i >> 4) & 63, rl = i & 15;
    Bsh[i] = ws[BM_OFF + k * 8192 + d * 128 + rb * 16 + rl];
  }
  if (t < 64)  gateL[t] = gate[t];
  if (t < 128) { invsL[t] = ws[INVS_OFF + t]; isgL[t] = ws[ISIG2_OFF + t]; }
  { const int k = t >> 7, n = t & 127; ssL[t] = ws[SK_OFF + k * Nn + nb * 128 + n]; }
  if (t < 32)  { const int k = t >> 4, r = t & 15; ccL[t] = ws[CK_OFF + k * 128 + rb * 16 + r]; }
  if (t < 16)  impL[t] = ws[IMP_OFF + rb * 16 + t];
  if (t < 6)   gamL[t] = ws[GAMMA_OFF + t];
  __syncthreads();

  const int lane = t & 31;
  const int half = lane >> 4;   // lanes 16..31
  const int l16  = lane & 15;
  const int wv   = t >> 5;      // wave id -> n sub-tile

  // ---- gaussian cross terms via f32 WMMA (16x16x4), K-loop over D=64 ----
  v8f acc0 = {0.f, 0.f, 0.f, 0.f, 0.f, 0.f, 0.f, 0.f};
  v8f acc1 = {0.f, 0.f, 0.f, 0.f, 0.f, 0.f, 0.f, 0.f};
  const float* zrow = &zs[(wv * 16 + l16) * 68];
  for (int d0 = 0; d0 < 64; d0 += 4) {
    const int ka = d0 + half * 2;  // A: lanes<16 hold K={d0,d0+1}, lanes>=16 K={d0+2,d0+3}
    v2f A;  A.x = zrow[ka];                    A.y = zrow[ka + 1];
    v2f B0; B0.x = Bsh[ka * 16 + l16];         B0.y = Bsh[(ka + 1) * 16 + l16];
    v2f B1; B1.x = Bsh[1024 + ka * 16 + l16];  B1.y = Bsh[1024 + (ka + 1) * 16 + l16];
    acc0 = __builtin_amdgcn_wmma_f32_16x16x4_f32(false, A, false, B0,
                                                 (short)0, acc0, false, false);
    acc1 = __builtin_amdgcn_wmma_f32_16x16x4_f32(false, A, false, B1,
                                                 (short)0, acc1, false, false);
  }

  // ---- nonlinear ops: lane owns the same 8 (n,r) cells as its WMMA acc ----
  const int nbase = wv * 16 + half * 8;  // C/D layout: vgpr j -> M = j + 8*half
  const float* arow = &as[l16 * 68];

  float nl[8][4];
#pragma unroll
  for (int j = 0; j < 8; ++j)
    { nl[j][0] = 0.f; nl[j][1] = 0.f; nl[j][2] = 0.f; nl[j][3] = 0.f; }
  unsigned int sgn = 0u;

  for (int d = 0; d < 64; ++d) {
    const float av = arow[d];
    const float g  = gateL[d];
    const float i2 = invsL[d],      i3 = invsL[64 + d];
    const float q4 = isgL[d],       q5 = isgL[64 + d];
#pragma unroll
    for (int j = 0; j < 8; ++j) {
      const float dist = zs[(nbase + j) * 68 + d] - av;
      // helmholtz k=2
      {
        float tt = dist * i2;
        float c  = __cosf(PI_F * tt);
        nl[j][0] += g * (__logf(fmaxf(fabsf(c), EPSF)) - fabsf(tt));
        sgn ^= (c < 0.f ? 1u : 0u) << (j * 4 + 0);
      }
      // helmholtz k=3
      {
        float tt = dist * i3;
        float c  = __cosf(PI_F * tt);
        nl[j][1] += g * (__logf(fmaxf(fabsf(c), EPSF)) - fabsf(tt));
        sgn ^= (c < 0.f ? 1u : 0u) << (j * 4 + 1);
      }
      // mexican hat k=4
      {
        float rr = dist * dist * q4;
        float f  = 1.f - rr;
        nl[j][2] += g * (__logf(fmaxf(fabsf(f), EPSF)) - 0.5f * rr);
        sgn ^= (f < 0.f ? 1u : 0u) << (j * 4 + 2);
      }
      // mexican hat k=5
      {
        float rr = dist * dist * q5;
        float f  = 1.f - rr;
        nl[j][3] += g * (__logf(fmaxf(fabsf(f), EPSF)) - 0.5f * rr);
        sgn ^= (f < 0.f ? 1u : 0u) << (j * 4 + 3);
      }
    }
  }

  // ---- combine and store ----
  const float cc0 = ccL[l16], cc1 = ccL[16 + l16];
  const float imp = impL[l16];
  const float g0 = gamL[0], g1 = gamL[1], g2 = gamL[2];
  const float g3 = gamL[3], g4 = gamL[4], g5 = gamL[5];
#pragma unroll
  for (int j = 0; j < 8; ++j) {
    const int nloc = nbase + j;
    const float lg0 = ssL[nloc] + acc0[j] + cc0;
    const float lg1 = ssL[128 + nloc] + acc1[j] + cc1;
    float res = g0 * __expf(lg0) + g1 * __expf(lg1);
    const float s2 = ((sgn >> (j * 4 + 0)) & 1u) ? -1.f : 1.f;
    const float s3 = ((sgn >> (j * 4 + 1)) & 1u) ? -1.f : 1.f;
    const float s4 = ((sgn >> (j * 4 + 2)) & 1u) ? -1.f : 1.f;
    const float s5 = ((sgn >> (j * 4 + 3)) & 1u) ? -1.f : 1.f;
    res += g2 * s2 * __expf(nl[j][0]);
    res += g3 * s3 * __expf(nl[j][1]);
    res += g4 * s4 * __expf(nl[j][2]);
    res += g5 * s5 * __expf(nl[j][3]);
    out[(nb * 128 + nloc) * Rr + rb * 16 + l16] = res * imp;
  }
}

extern "C" void kernel_launch(void* const* d_in, const int* in_sizes, int n_in,
                              void* d_out, int out_size, void* d_ws, size_t ws_size,
                              hipStream_t stream) {
  const float* z    = (const float*)d_in[0];
  const float* gate = (const float*)d_in[1];
  const float* anch = (const float*)d_in[2];
  const float* rimp = (const float*)d_in[3];
  const float* lsig = (const float*)d_in[4];
  const float* lgam = (const float*)d_in[5];
  float* ws  = (float*)d_ws;
  float* out = (float*)d_out;

  hipLaunchKernelGGL(prep_const_kernel, dim3(1), dim3(256), 0, stream,
                     gate, anch, rimp, lsig, lgam, ws);
  hipLaunchKernelGGL(prep_sz_kernel, dim3(8), dim3(256), 0, stream, z, ws);
  hipLaunchKernelGGL(phi_main_kernel, dim3(16, 8), dim3(256), 0, stream,
                     z, gate, anch, ws, out);
}